// Canvas_by_Distance_4346506904131
// MI455X (gfx1250) — compile-verified
//
#include <hip/hip_runtime.h>

typedef __attribute__((ext_vector_type(2))) float v2f;
typedef __attribute__((ext_vector_type(4))) float v4f;
typedef __attribute__((ext_vector_type(8))) float v8f;

#define CANVAS     1024
#define IM         4096
#define NCOL       16
#define PLANE_IN   (CANVAS * CANVAS)   // 1048576
#define PLANE_OUT  (IM * IM)           // 16777216 (fits in 32-bit index)
#define GROUPS_PER_WAVE 8

__device__ __forceinline__ float fast_sigmoid(float x) {
    // v_exp_f32 + v_add + v_rcp_f32 (no IEEE div expansion)
    return __builtin_amdgcn_rcpf(1.0f + __expf(-x));
}

// One wave32 classifies 16 pixels per V_WMMA_F32_16X16X4_F32:
//   D[m][n] = (-2*palette[m]) . sigmoid(w[n]) + |palette[m]|^2 + 8
// argmax_m D[.][n] == argmax_m squared-distance (per-column |w_n|^2 and the
// +8 bias are constant in m). The +8 makes every D strictly positive so the
// packed-uint max trick below is order-preserving.
__global__ __launch_bounds__(256)
void canvas_by_distance_kernel(const float* __restrict__ weight,
                               const float* __restrict__ palette,
                               float* __restrict__ out) {
    const int lane = threadIdx.x & 31;   // wave32
    const int half = lane >> 4;          // 0: lanes 0-15, 1: lanes 16-31
    const int lm   = lane & 15;
    const int wave_id = blockIdx.x * (blockDim.x >> 5) + (threadIdx.x >> 5);

    // ---- per-wave constants -------------------------------------------------
    // A matrix 16x4 (MxK): lane m -> {K0,K1}; lane m+16 -> {K2,K3}
    const float p0 = palette[lm * 3 + 0];
    const float p1 = palette[lm * 3 + 1];
    const float p2 = palette[lm * 3 + 2];
    v2f amat;
    amat.x = half ? (-2.0f * p2) : (-2.0f * p0);
    amat.y = half ? 0.0f         : (-2.0f * p1);

    // C bias: |palette[m]|^2 + 8. Layout: VGPR v -> M=v (lower), M=v+8 (upper).
    float nrm[NCOL];
#pragma unroll
    for (int m = 0; m < NCOL; ++m) {
        const float r = palette[m * 3 + 0];
        const float g = palette[m * 3 + 1];
        const float b = palette[m * 3 + 2];
        nrm[m] = r * r + g * g + b * b + 8.0f;
    }
    v8f cbias;
#pragma unroll
    for (int v = 0; v < 8; ++v) cbias[v] = half ? nrm[v + 8] : nrm[v];

    // Packed-key index field: store (15 - m) in the low 4 bits so that on a
    // value tie the LARGER key has the SMALLER color index (jnp.argmax picks
    // the first match).
    const int idx_base = 15 - half * 8;                 // m = half*8 + v
    const unsigned woff0 = half ? 2u * PLANE_IN : 0u;   // r-plane / b-plane

    // ---- per-group work -----------------------------------------------------
    const int g0 = wave_id * GROUPS_PER_WAVE;
#pragma unroll 1
    for (int it = 0; it < GROUPS_PER_WAVE; ++it) {
        const int g    = g0 + it;
        const int cy   = g >> 6;              // 64 groups of 16 pixels per row
        const int cx0  = (g & 63) << 4;
        const int pix  = cx0 + lm;
        const unsigned base = (unsigned)(cy * CANVAS + pix);

        // B matrix 4x16 (KxN): lane n -> {sig_r, sig_g}; lane n+16 -> {sig_b, 0}
        const float w0 = __builtin_nontemporal_load(weight + base + woff0);
        const float w1 = __builtin_nontemporal_load(weight + base + PLANE_IN);
        v2f bmat;
        bmat.x = fast_sigmoid(w0);
        bmat.y = half ? 0.0f : fast_sigmoid(w1);

        // D = A x B + C  (EXEC all-ones: no divergent control flow anywhere)
        v8f d = __builtin_amdgcn_wmma_f32_16x16x4_f32(
            /*neg_a=*/false, amat, /*neg_b=*/false, bmat,
            /*c_mod=*/(short)0, cbias, /*reuse_a=*/false, /*reuse_b=*/false);

        // Lane-local argmax via packed unsigned max: value in bits [31:4],
        // complemented index in [3:0]. All D > 0 so uint order == float order.
        unsigned key = (__float_as_uint(d[0]) & 0xFFFFFFF0u) | (unsigned)idx_base;
#pragma unroll
        for (int v = 1; v < 8; ++v) {
            const unsigned kv =
                (__float_as_uint(d[v]) & 0xFFFFFFF0u) | (unsigned)(idx_base - v);
            key = key > kv ? key : kv;        // v_max_u32 / v_max3_u32
        }
        // Symmetric cross-half merge: lane n <-> n+16 share a pixel column, so
        // after one xor-shuffle + max BOTH halves hold the winner (no broadcast).
        const unsigned okey = (unsigned)__shfl_xor((int)key, 16, 32);
        key = key > okey ? key : okey;
        const int idx = 15 - (int)(key & 15u);

        const float cr = palette[idx * 3 + 0];
        const float cg = palette[idx * 3 + 1];
        const float cb = palette[idx * 3 + 2];
        const v4f qr = {cr, cr, cr, cr};
        const v4f qg = {cg, cg, cg, cg};
        const v4f qb = {cb, cb, cb, cb};

        // 4x nearest upsample. Lane L owns output columns [4*pix, 4*pix+4) in
        // sub-rows {half, half+2}: the two halves cover rows 0..3 across the
        // two stores per plane -> 6 coalesced global_store_b128, NT hint.
        // Single uniform base + unsigned 32-bit element offsets (max 48M < 2^32)
        // so the backend can use the saddr form (like it already does for the
        // weight loads) instead of per-iteration 64-bit carry chains; the
        // +2*IM row delta (32768 B) folds into the 24-bit instruction offset.
        const unsigned base0 = (unsigned)((4 * cy + half) * IM + 4 * pix);
        const unsigned o0 = base0;
        const unsigned o1 = base0 + (unsigned)PLANE_OUT;
        const unsigned o2 = base0 + 2u * (unsigned)PLANE_OUT;
        __builtin_nontemporal_store(qr, (v4f*)(out + o0));
        __builtin_nontemporal_store(qr, (v4f*)(out + o0 + 2 * IM));
        __builtin_nontemporal_store(qg, (v4f*)(out + o1));
        __builtin_nontemporal_store(qg, (v4f*)(out + o1 + 2 * IM));
        __builtin_nontemporal_store(qb, (v4f*)(out + o2));
        __builtin_nontemporal_store(qb, (v4f*)(out + o2 + 2 * IM));
    }
}

extern "C" void kernel_launch(void* const* d_in, const int* in_sizes, int n_in,
                              void* d_out, int out_size, void* d_ws, size_t ws_size,
                              hipStream_t stream) {
    const float* weight  = (const float*)d_in[0];   // (3, 1024, 1024) f32
    const float* palette = (const float*)d_in[1];   // (16, 3) f32
    float*       out     = (float*)d_out;           // (3, 4096, 4096) f32

    const int groups  = (CANVAS * CANVAS) / 16;          // 65536
    const int waves   = groups / GROUPS_PER_WAVE;        // 8192
    const int threads = 256;                             // 8 waves per block
    const int blocks  = waves * 32 / threads;            // 1024

    canvas_by_distance_kernel<<<blocks, threads, 0, stream>>>(weight, palette, out);
}